// PositionCRFBatch_43508018709172
// MI455X (gfx1250) — compile-verified
//
#include <hip/hip_runtime.h>
#include <math.h>

#define B_ 64
#define L_ 1024
#define H_ 768
#define K_ 4

typedef float v2f __attribute__((ext_vector_type(2)));
typedef float v8f __attribute__((ext_vector_type(8)));

// ---------------------------------------------------------------------------
// Kernel 0: zero-padded copy of w: wpad[16][768], rows 0-3 = w, rows 4-15 = 0.
// Lets the WMMA B-fragment load directly with no per-iteration masking VALU.
// ---------------------------------------------------------------------------
__global__ void pad_w_kernel(const float* __restrict__ w,
                             float* __restrict__ wpad) {
  int i = blockIdx.x * blockDim.x + threadIdx.x;   // 16*768 = 12288 elements
  if (i < 16 * H_) wpad[i] = (i < K_ * H_) ? w[i] : 0.0f;
}

// ---------------------------------------------------------------------------
// Kernel 1: feats[B*L, 4] = hidden[B*L, 768] @ w[4,768]^T + bias
// One wave per 16-row tile of (B*L), V_WMMA_F32_16X16X4_F32, two independent
// accumulator chains (even/odd K-chunks) to double intra-wave ILP.
// A layout (ISA 7.12.2, 32-bit A 16x4): lanes 0-15 hold K=0,1 in VGPR0,1;
// lanes 16-31 hold K=2,3. B mirrors with lane%16 = N (class, zero-padded
// 4 -> 16 via wpad). D: VGPR r -> row M = r + 8*(lane/16), N = lane%16.
// ---------------------------------------------------------------------------
__global__ void feats_wmma_kernel(const float* __restrict__ hidden,
                                  const float* __restrict__ wpad,
                                  const float* __restrict__ bias,
                                  float* __restrict__ feats) {
  const int wave = (int)((blockIdx.x * blockDim.x + threadIdx.x) >> 5);
  const int lane = threadIdx.x & 31;
  const int half = lane >> 4;
  const int m    = lane & 15;         // A: row-in-tile ; B/D: column N
  const long long row0 = (long long)wave * 16;
  const int n = m;                    // output class column (padded to 16)

  const float* arow = hidden + (row0 + m) * (long long)H_ + 2 * half;
  const float* brow = wpad + (long long)n * H_ + 2 * half;

  v8f c0 = {0.f, 0.f, 0.f, 0.f, 0.f, 0.f, 0.f, 0.f};
  v8f c1 = {0.f, 0.f, 0.f, 0.f, 0.f, 0.f, 0.f, 0.f};
#pragma unroll 4
  for (int kk = 0; kk < H_; kk += 8) {
    v2f a0 = *(const v2f*)(arow + kk);
    v2f b0 = *(const v2f*)(brow + kk);
    v2f a1 = *(const v2f*)(arow + kk + 4);
    v2f b1 = *(const v2f*)(brow + kk + 4);
    // 8 args: (neg_a, A, neg_b, B, c_mod, C, reuse_a, reuse_b)
    c0 = __builtin_amdgcn_wmma_f32_16x16x4_f32(false, a0, false, b0,
                                               (short)0, c0, false, false);
    c1 = __builtin_amdgcn_wmma_f32_16x16x4_f32(false, a1, false, b1,
                                               (short)0, c1, false, false);
  }

  if (n < K_) {
    const float bn = bias[n];
#pragma unroll
    for (int r = 0; r < 8; ++r) {
      long long row = row0 + r + 8 * half;
      feats[row * K_ + n] = (c0[r] + c1[r]) + bn;
    }
  }
}

// ---------------------------------------------------------------------------
// Kernel 2: CRF forward-backward, one thread per batch element (64 threads).
// K=4 fully unrolled in registers. hidden_mask==0 means VALID position.
// Stores alphas (for marginals), sp = class-1 marginal, gamma = sum(sp)/2.
// ---------------------------------------------------------------------------
__device__ __forceinline__ float lse4(float x0, float x1, float x2, float x3) {
  float mx = fmaxf(fmaxf(x0, x1), fmaxf(x2, x3));
  float s = __expf(x0 - mx) + __expf(x1 - mx) + __expf(x2 - mx) + __expf(x3 - mx);
  return mx + __logf(s);
}

__global__ void crf_kernel(const float* __restrict__ feats,
                           const int* __restrict__ hmask,
                           const float* __restrict__ trans,
                           float* __restrict__ alphas,
                           float* __restrict__ sp,
                           float* __restrict__ gamma) {
  const int b = blockIdx.x * blockDim.x + threadIdx.x;
  if (b >= B_) return;

  float T[16];
#pragma unroll
  for (int i = 0; i < 16; ++i) T[i] = trans[i];

  const float* f  = feats + (long long)b * L_ * K_;
  const int*   hm = hmask + b * L_;
  float4* al = (float4*)(alphas + (long long)b * L_ * K_);

  // ---- forward ----
  float a0 = f[0], a1 = f[1], a2 = f[2], a3 = f[3];
  al[0] = make_float4(a0, a1, a2, a3);
  for (int l = 1; l < L_; ++l) {
    if (hm[l] == 0) {
      const float* fl = f + l * K_;
      float n0 = fl[0] + lse4(a0 + T[0], a1 + T[4], a2 + T[8],  a3 + T[12]);
      float n1 = fl[1] + lse4(a0 + T[1], a1 + T[5], a2 + T[9],  a3 + T[13]);
      float n2 = fl[2] + lse4(a0 + T[2], a1 + T[6], a2 + T[10], a3 + T[14]);
      float n3 = fl[3] + lse4(a0 + T[3], a1 + T[7], a2 + T[11], a3 + T[15]);
      a0 = n0; a1 = n1; a2 = n2; a3 = n3;
    }
    al[l] = make_float4(a0, a1, a2, a3);
  }
  const float logZ = lse4(a0, a1, a2, a3);

  // ---- backward + class-1 marginal ----
  float b0 = 0.f, b1 = 0.f, b2 = 0.f, b3 = 0.f;
  float sum_sp = 0.f;
  float* spb = sp + b * L_;
  for (int l = L_ - 1; l >= 0; --l) {
    float4 av = al[l];
    float valid = (hm[l] == 0) ? 1.0f : 0.0f;
    float s = __expf(av.y + b1 - logZ) * valid;
    spb[l] = s;
    sum_sp += s;
    if (l > 0 && hm[l] == 0) {
      const float* fl = f + l * K_;
      float t0 = fl[0] + b0, t1 = fl[1] + b1, t2 = fl[2] + b2, t3 = fl[3] + b3;
      float n0 = lse4(T[0]  + t0, T[1]  + t1, T[2]  + t2, T[3]  + t3);
      float n1 = lse4(T[4]  + t0, T[5]  + t1, T[6]  + t2, T[7]  + t3);
      float n2 = lse4(T[8]  + t0, T[9]  + t1, T[10] + t2, T[11] + t3);
      float n3 = lse4(T[12] + t0, T[13] + t1, T[14] + t2, T[15] + t3);
      b0 = n0; b1 = n1; b2 = n2; b3 = n3;
    }
  }
  gamma[b] = 0.5f * sum_sp;
}

// ---------------------------------------------------------------------------
// Kernel 3: sent_vs[b,h] = (sum_l sp[b,l] * hidden[b,l,h]) / gamma[b]
// One block per (b, 256-wide h chunk); coalesced over h, sp broadcast.
// ---------------------------------------------------------------------------
__global__ void sentvs_kernel(const float* __restrict__ hidden,
                              const float* __restrict__ sp,
                              const float* __restrict__ gamma,
                              float* __restrict__ out_sent) {
  const int chunks = H_ / 256;                 // 3
  const int b = blockIdx.x / chunks;
  const int h = (blockIdx.x % chunks) * 256 + threadIdx.x;
  const float* hb  = hidden + (long long)b * L_ * H_ + h;
  const float* spb = sp + b * L_;
  float acc = 0.f;
  for (int l = 0; l < L_; ++l) {
    __builtin_prefetch(hb + (long long)(l + 16) * H_, 0, 1);
    acc += spb[l] * hb[(long long)l * H_];
  }
  out_sent[b * H_ + h] = acc / gamma[b];
}

// ---------------------------------------------------------------------------
// Kernel 4: s_prob = sp / gamma[b]
// ---------------------------------------------------------------------------
__global__ void sprob_kernel(const float* __restrict__ sp,
                             const float* __restrict__ gamma,
                             float* __restrict__ out_sprob) {
  int i = blockIdx.x * blockDim.x + threadIdx.x;
  if (i < B_ * L_) out_sprob[i] = sp[i] / gamma[i / L_];
}

// ---------------------------------------------------------------------------
// Kernel 5: norm_pen = C1*(relu(T10-T00)+relu(T01-T11)) + C2*mean_b(sum_l|sp/g|)
// ---------------------------------------------------------------------------
__global__ void pen_kernel(const float* __restrict__ sp,
                           const float* __restrict__ gamma,
                           const float* __restrict__ trans,
                           float* __restrict__ out_pen) {
  __shared__ float red[B_];
  const int t = threadIdx.x;              // one thread per batch element
  const float inv_g = 1.0f / gamma[t];
  float s = 0.f;
  const float* spb = sp + t * L_;
  for (int l = 0; l < L_; ++l) s += fabsf(spb[l] * inv_g);
  red[t] = s;
  __syncthreads();
  if (t == 0) {
    float tot = 0.f;
    for (int i = 0; i < B_; ++i) tot += red[i];
    // T[1,0]=trans[4], T[0,0]=trans[0], T[0,1]=trans[1], T[1,1]=trans[5]
    float pena = fmaxf(trans[4] - trans[0], 0.f) + fmaxf(trans[1] - trans[5], 0.f);
    out_pen[0] = 0.1f * pena + 0.1f * (tot / (float)B_);
  }
}

// ---------------------------------------------------------------------------
extern "C" void kernel_launch(void* const* d_in, const int* in_sizes, int n_in,
                              void* d_out, int out_size, void* d_ws, size_t ws_size,
                              hipStream_t stream) {
  const float* hidden = (const float*)d_in[0];
  const int*   hmask  = (const int*)d_in[1];
  const float* w      = (const float*)d_in[7];
  const float* bias   = (const float*)d_in[8];
  const float* trans  = (const float*)d_in[9];

  float* ws     = (float*)d_ws;
  float* feats  = ws;                        // B*L*K  = 262144 floats
  float* alphas = ws + 262144;               // B*L*K  = 262144 floats
  float* sp     = ws + 524288;               // B*L    =  65536 floats
  float* gamma  = ws + 589824;               // B      =     64 floats
  float* wpad   = ws + 589888;               // 16*H   =  12288 floats

  float* out_sent  = (float*)d_out;          // [B,H]
  float* out_pen   = out_sent + B_ * H_;     // scalar
  float* out_sprob = out_pen + 1;            // [B,L]

  // 0) zero-padded w (16 x 768)
  pad_w_kernel<<<(16 * H_ + 255) / 256, 256, 0, stream>>>(w, wpad);

  // 1) feats GEMM via f32 WMMA: 4096 row-tiles, 8 waves per 256-thread block
  feats_wmma_kernel<<<(B_ * L_ / 16) / 8, 256, 0, stream>>>(hidden, wpad, bias, feats);

  // 2) CRF forward-backward: one thread per batch element
  crf_kernel<<<1, B_, 0, stream>>>(feats, hmask, trans, alphas, sp, gamma);

  // 3) weighted hidden reduction -> sent_vs
  sentvs_kernel<<<B_ * (H_ / 256), 256, 0, stream>>>(hidden, sp, gamma, out_sent);

  // 4) s_prob
  sprob_kernel<<<(B_ * L_) / 256, 256, 0, stream>>>(sp, gamma, out_sprob);

  // 5) scalar penalty
  pen_kernel<<<1, B_, 0, stream>>>(sp, gamma, trans, out_pen);
}